// SimpleSelfAttention_8538394984938
// MI455X (gfx1250) — compile-verified
//
#include <hip/hip_runtime.h>
#include <math.h>

// Self-attention y = softmax(x @ x^T) @ x for x:[B,4096,64] fp32.
// Compute-bound; fp32 WMMA (v_wmma_f32_16x16x4_f32) flash attention.
// One 16-row query tile per wave32; 64 keys per online-softmax update to
// amortize cross-lane reductions; row-sums computed by P @ ones WMMA so the
// l accumulator arrives broadcast in C-layout (no shuffle tree for sums).

typedef float v2f __attribute__((ext_vector_type(2)));
typedef float v8f __attribute__((ext_vector_type(8)));

#define S_LEN 4096
#define D_DIM 64
#define QTILE 16
#define KTILE 64                 // keys per softmax update = 4 WMMA N-tiles
#define WAVES_PER_BLOCK 8

__global__ __launch_bounds__(256)
void attn_wmma_f32_kernel(const float* __restrict__ x, float* __restrict__ y) {
  // Per-wave 16x64 staging buffer: P from C-layout -> A-layout. 4KB/wave.
  __shared__ float pbuf[WAVES_PER_BLOCK][QTILE * KTILE];

  const int lane = threadIdx.x & 31;
  const int wave = threadIdx.x >> 5;
  const int tile_id = blockIdx.x * WAVES_PER_BLOCK + wave;
  const int qtiles_per_batch = S_LEN / QTILE;          // 256
  const int b  = tile_id / qtiles_per_batch;
  const int qt = tile_id % qtiles_per_batch;

  const int m     = lane & 15;   // A-layout row M / B,C-layout column N
  const int half  = lane >> 4;
  const int khalf = half * 2;    // lanes 16-31 hold K=2,3 of each K=4 chunk

  const float* xb = x + (size_t)b * S_LEN * D_DIM;

  // ---- Q tile (16x64) resident in A-layout registers: 16 chunks of K=4.
  v2f aq[16];
  {
    const float* qrow = xb + (size_t)(qt * QTILE + m) * D_DIM;
    #pragma unroll
    for (int c = 0; c < 16; ++c) {
      aq[c].x = qrow[4 * c + khalf];
      aq[c].y = qrow[4 * c + khalf + 1];
    }
  }

  // Online-softmax state. Lane owns rows r + 8*half (replicated across its
  // 16-lane half, matching the WMMA C/D layout).
  float m_run[8];
  #pragma unroll
  for (int r = 0; r < 8; ++r) m_run[r] = -INFINITY;
  v8f lacc = {};                 // running row-sums (C-layout, via P @ ones)
  v8f o[4] = {};                 // O accumulator: 16x64 as four 16x16 C tiles

  const v2f ones = {1.0f, 1.0f}; // B operand of all-ones for row sums

  for (int kt = 0; kt < S_LEN / KTILE; ++kt) {
    const int k0 = kt * KTILE;

    // ---- S = Q @ K^T for 4 key sub-tiles (16 keys each).
    v8f s[4] = {};
    #pragma unroll
    for (int t = 0; t < 4; ++t) {
      const float* krow = xb + (size_t)(k0 + t * 16 + m) * D_DIM;
      #pragma unroll
      for (int c = 0; c < 16; ++c) {
        v2f bk;
        bk.x = krow[4 * c + khalf];
        bk.y = krow[4 * c + khalf + 1];
        s[t] = __builtin_amdgcn_wmma_f32_16x16x4_f32(
            false, aq[c], false, bk, (short)0, s[t], false, false);
      }
    }

    // ---- Online softmax over 64 keys: local 4-way max, then ONE shuffle tree.
    float alpha[8];
    #pragma unroll
    for (int r = 0; r < 8; ++r) {
      float v = fmaxf(fmaxf(s[0][r], s[1][r]), fmaxf(s[2][r], s[3][r]));
      #pragma unroll
      for (int mask = 1; mask < 16; mask <<= 1)
        v = fmaxf(v, __shfl_xor(v, mask, 32));
      const float mn = fmaxf(m_run[r], v);
      alpha[r] = __expf(m_run[r] - mn);
      m_run[r] = mn;
      #pragma unroll
      for (int t = 0; t < 4; ++t)
        s[t][r] = __expf(s[t][r] - mn);   // s now holds P (C-layout)
    }

    // Rescale running accumulators by alpha.
    #pragma unroll
    for (int r = 0; r < 8; ++r) lacc[r] *= alpha[r];
    #pragma unroll
    for (int n = 0; n < 4; ++n)
      #pragma unroll
      for (int r = 0; r < 8; ++r)
        o[n][r] *= alpha[r];

    // ---- P (16x64): C-layout -> A-layout through LDS (same wave, in-order).
    float* pw = pbuf[wave];
    #pragma unroll
    for (int t = 0; t < 4; ++t)
      #pragma unroll
      for (int r = 0; r < 8; ++r)
        pw[(r + 8 * half) * KTILE + t * 16 + m] = s[t][r];
    __builtin_amdgcn_wave_barrier();
    v2f ap[16];
    #pragma unroll
    for (int c = 0; c < 16; ++c) {
      ap[c].x = pw[m * KTILE + 4 * c + khalf];
      ap[c].y = pw[m * KTILE + 4 * c + khalf + 1];
    }

    // ---- Row sums: lacc += P @ ones (lands broadcast in C-layout).
    #pragma unroll
    for (int c = 0; c < 16; ++c)
      lacc = __builtin_amdgcn_wmma_f32_16x16x4_f32(
          false, ap[c], false, ones, (short)0, lacc, false, false);

    // ---- O += P @ V (V = 64x64 key tile, row-major -> natural B layout).
    const float* vbase = xb + (size_t)k0 * D_DIM;
    #pragma unroll
    for (int n = 0; n < 4; ++n) {
      #pragma unroll
      for (int c = 0; c < 16; ++c) {
        v2f bv;
        bv.x = vbase[(4 * c + khalf)     * D_DIM + n * 16 + m];
        bv.y = vbase[(4 * c + khalf + 1) * D_DIM + n * 16 + m];
        o[n] = __builtin_amdgcn_wmma_f32_16x16x4_f32(
            false, ap[c], false, bv, (short)0, o[n], false, false);
      }
    }
  }

  // ---- Epilogue: y = O / l (row r + 8*half; coalesced 16-lane stores).
  float inv[8];
  #pragma unroll
  for (int r = 0; r < 8; ++r) inv[r] = 1.0f / lacc[r];
  float* yb = y + ((size_t)b * S_LEN + (size_t)qt * QTILE) * D_DIM;
  #pragma unroll
  for (int n = 0; n < 4; ++n) {
    #pragma unroll
    for (int r = 0; r < 8; ++r) {
      const int row = r + 8 * half;
      yb[row * D_DIM + n * 16 + m] = o[n][r] * inv[r];
    }
  }
}

extern "C" void kernel_launch(void* const* d_in, const int* in_sizes, int n_in,
                              void* d_out, int out_size, void* d_ws, size_t ws_size,
                              hipStream_t stream) {
  const float* x = (const float*)d_in[0];
  float* y = (float*)d_out;
  const int B = in_sizes[0] / (S_LEN * D_DIM);          // 4
  const int tiles = B * (S_LEN / QTILE);                // 1024 query tiles
  const int blocks = tiles / WAVES_PER_BLOCK;           // 128 blocks of 8 waves
  attn_wmma_f32_kernel<<<blocks, 256, 0, stream>>>(x, y);
}